// BatchNeuralKB_37701222924640
// MI455X (gfx1250) — compile-verified
//
#include <hip/hip_runtime.h>
#include <math.h>

typedef __attribute__((ext_vector_type(2))) float v2f;
typedef __attribute__((ext_vector_type(8))) float v8f;

#define TILE_F     16
#define LDS_STRIDE 132   // 128 + 4 pad floats -> column reads hit distinct LDS banks

// ---------------------------------------------------------------------------
// Pre-pass: per-(b,f) fixed distance constants
//   cfix3[b,f] = ||rel-f1||^2 + ||arg1-f2||^2 + ||f3||^2   (+1e30 if f>=nbf)
//   cfix2[b,f] = ||rel-f1||^2 + ||arg2-f3||^2 + ||f2||^2   (+1e30 if f>=nbf)
// ---------------------------------------------------------------------------
__device__ __forceinline__ float sqd4(float4 a, float4 b) {
    float x = a.x - b.x, y = a.y - b.y, z = a.z - b.z, w = a.w - b.w;
    return x * x + y * y + z * z + w * w;
}
__device__ __forceinline__ float nrm4(float4 a) {
    return a.x * a.x + a.y * a.y + a.z * a.z + a.w * a.w;
}

__global__ __launch_bounds__(256) void cfix_kernel(
    const float* __restrict__ rel, const float* __restrict__ arg1,
    const float* __restrict__ arg2, const float* __restrict__ f1,
    const float* __restrict__ f2, const float* __restrict__ f3,
    const long long* __restrict__ nb_facts,
    float* __restrict__ cfix3, float* __restrict__ cfix2,
    int B, int F, int E)
{
    int idx = blockIdx.x * blockDim.x + threadIdx.x;
    if (idx >= B * F) return;
    int b = idx / F;
    int f = idx - b * F;

    const float4* r4  = (const float4*)(rel  + (size_t)b * E);
    const float4* a14 = (const float4*)(arg1 + (size_t)b * E);
    const float4* a24 = (const float4*)(arg2 + (size_t)b * E);
    const float4* p1  = (const float4*)(f1 + ((size_t)b * F + f) * E);
    const float4* p2  = (const float4*)(f2 + ((size_t)b * F + f) * E);
    const float4* p3  = (const float4*)(f3 + ((size_t)b * F + f) * E);

    float d_rel = 0.f, d_a1 = 0.f, d_a2 = 0.f, sq2 = 0.f, sq3 = 0.f;
    for (int i = 0; i < E / 4; ++i) {
        float4 rv = r4[i], av1 = a14[i], av2 = a24[i];
        float4 v1 = p1[i], v2 = p2[i], v3 = p3[i];
        d_rel += sqd4(rv, v1);
        d_a1  += sqd4(av1, v2);
        d_a2  += sqd4(av2, v3);
        sq2   += nrm4(v2);
        sq3   += nrm4(v3);
    }
    float maskpen = (f < (int)nb_facts[b]) ? 0.f : 1.0e30f;
    cfix3[(size_t)b * F + f] = d_rel + d_a1 + sq3 + maskpen;  // score_sp branch
    cfix2[(size_t)b * F + f] = d_rel + d_a2 + sq2 + maskpen;  // score_po branch
}

// ---------------------------------------------------------------------------
// Main fused kernel. Per workgroup (8 wave32s): 128 entity rows, full F loop.
// fact3/fact2 tiles are staged cooperatively in double-buffered, bank-padded
// LDS; global prefetch of tile t+1 overlaps the 64-WMMA compute of tile t.
// Each wave holds its 16x128 A tile resident in 64 VGPRs and runs four
// independent f32 WMMA accumulator chains, with a fused running-min epilogue:
//   score[b,n] = (n<nbe) ? exp(-(min_f (cfix[f] - 2*ent.fX) + ||ent||^2)/2) : 0
// ---------------------------------------------------------------------------
__global__ __launch_bounds__(256) void score_kernel(
    const float* __restrict__ ent,
    const float* __restrict__ f2, const float* __restrict__ f3,
    const float* __restrict__ cfix3, const float* __restrict__ cfix2,
    const long long* __restrict__ nb_facts,
    const long long* __restrict__ nb_entities,
    float* __restrict__ out,
    int B, int N, int F, int E)
{
    __shared__ __align__(16) float ldsF3[2][TILE_F * LDS_STRIDE];
    __shared__ __align__(16) float ldsF2[2][TILE_F * LDS_STRIDE];

    const int tilesPerB = N / 128;            // 128 entity rows per workgroup
    int b    = blockIdx.x / tilesPerB;
    int tile = blockIdx.x - b * tilesPerB;

    int lane = threadIdx.x & 31;
    int wave = threadIdx.x >> 5;
    int hi   = lane >> 4;                      // k-half selector (ISA A/B layout)
    int lo   = lane & 15;                      // row (A) / fact column (B) index
    int rowBase = tile * 128 + wave * 16;

    // Cooperative-copy mapping: thread -> (fact row, 8-float slice of E)
    int cpF = threadIdx.x >> 4;                // 0..15
    int cpE = (threadIdx.x & 15) * 8;          // 0,8,...,120
    const float* g3 = f3 + ((size_t)b * F + cpF) * E + cpE;
    const float* g2 = f2 + ((size_t)b * F + cpF) * E + cpE;
    float* l3 = &ldsF3[0][0] ;                 // buffer base; index per tile below
    float* l2 = &ldsF2[0][0] ;
    int ldsOff = cpF * LDS_STRIDE + cpE;

    // --- Load A tile: 16 entity rows x 128 E, resident (64 VGPRs/lane) ---
    // ISA 32-bit A 16x4 layout: lanes 0-15 hold K={k0,k0+1}, lanes 16-31 hold
    // K={k0+2,k0+3} for row M = lane&15.
    v2f a_reg[32];
    float entsq_part = 0.f;
    const float* arow = ent + ((size_t)b * N + rowBase + lo) * E + 2 * hi;
#pragma unroll
    for (int kk = 0; kk < 32; ++kk) {
        v2f av = *(const v2f*)(arow + 4 * kk);
        a_reg[kk] = av;
        entsq_part += av.x * av.x + av.y * av.y;
    }
    float entsq = entsq_part + __shfl_xor(entsq_part, 16, 32);

    int nbf  = (int)nb_facts[b];
    int fEnd = (nbf + 15) & ~15;
    if (fEnd > F) fEnd = F;
    int ntiles = fEnd / TILE_F;                // >= 1 (nb_facts >= 1)

    v8f m3, m2;                                // running min accumulators
#pragma unroll
    for (int r = 0; r < 8; ++r) { m3[r] = 3.0e38f; m2[r] = 3.0e38f; }

    // Prologue: stage tile 0 into buffer 0
    {
        const float4* s3 = (const float4*)g3;
        const float4* s2 = (const float4*)g2;
        float4 q0 = s3[0], q1 = s3[1], q2 = s2[0], q3 = s2[1];
        float4* d3 = (float4*)&ldsF3[0][ldsOff];
        float4* d2 = (float4*)&ldsF2[0][ldsOff];
        d3[0] = q0; d3[1] = q1; d2[0] = q2; d2[1] = q3;
    }
    __syncthreads();

    for (int t = 0; t < ntiles; ++t) {
        int cur = t & 1;
        int fbase = t * TILE_F;
        bool hasNext = (t + 1) < ntiles;

        // Issue global prefetch for tile t+1 (loads fly during compute)
        float4 p0, p1, p2, p3;
        if (hasNext) {
            size_t goff = (size_t)(fbase + TILE_F) * E;
            const float4* s3 = (const float4*)(g3 + goff);
            const float4* s2 = (const float4*)(g2 + goff);
            p0 = s3[0]; p1 = s3[1]; p2 = s2[0]; p3 = s2[1];
        }

        // ---- Compute: 64 f32 WMMAs on the LDS-resident tile ----
        v8f c3e = {}, c3o = {}, c2e = {}, c2o = {};   // 4 independent chains
        const float* L3 = &ldsF3[cur][lo * LDS_STRIDE + 2 * hi];
        const float* L2 = &ldsF2[cur][lo * LDS_STRIDE + 2 * hi];
#pragma unroll
        for (int g = 0; g < 4; ++g) {
            v2f b3r[8], b2r[8];                // distinct regs -> clausable loads
#pragma unroll
            for (int i = 0; i < 8; ++i) {
                b3r[i] = *(const v2f*)(L3 + 4 * (g * 8 + i));
                b2r[i] = *(const v2f*)(L2 + 4 * (g * 8 + i));
            }
#pragma unroll
            for (int i = 0; i < 8; i += 2) {
                c3e = __builtin_amdgcn_wmma_f32_16x16x4_f32(
                    false, a_reg[g * 8 + i], false, b3r[i], (short)0, c3e, false, false);
                c2e = __builtin_amdgcn_wmma_f32_16x16x4_f32(
                    false, a_reg[g * 8 + i], false, b2r[i], (short)0, c2e, false, false);
                c3o = __builtin_amdgcn_wmma_f32_16x16x4_f32(
                    false, a_reg[g * 8 + i + 1], false, b3r[i + 1], (short)0, c3o, false, false);
                c2o = __builtin_amdgcn_wmma_f32_16x16x4_f32(
                    false, a_reg[g * 8 + i + 1], false, b2r[i + 1], (short)0, c2o, false, false);
            }
        }

        // ---- Fused epilogue: dist = cfix[f] - 2*cross ; running min ----
        float cf3 = cfix3[(size_t)b * F + fbase + lo];
        float cf2 = cfix2[(size_t)b * F + fbase + lo];
#pragma unroll
        for (int r = 0; r < 8; ++r) {
            m3[r] = fminf(m3[r], cf3 - 2.0f * (c3e[r] + c3o[r]));
            m2[r] = fminf(m2[r], cf2 - 2.0f * (c2e[r] + c2o[r]));
        }

        // Drain prefetch into the other LDS buffer, then sync
        if (hasNext) {
            float4* d3 = (float4*)&ldsF3[cur ^ 1][ldsOff];
            float4* d2 = (float4*)&ldsF2[cur ^ 1][ldsOff];
            d3[0] = p0; d3[1] = p1; d2[0] = p2; d2[1] = p3;
        }
        __syncthreads();
    }

    // --- Min across the 16 fact columns (lanes within each 16-lane half) ---
#pragma unroll
    for (int r = 0; r < 8; ++r) {
        float v3 = m3[r], v2v = m2[r];
#pragma unroll
        for (int m = 1; m <= 8; m <<= 1) {
            v3  = fminf(v3,  __shfl_xor(v3,  m, 32));
            v2v = fminf(v2v, __shfl_xor(v2v, m, 32));
        }
        m3[r] = v3;
        m2[r] = v2v;
    }

    // C/D layout: VGPR r, lanes 0-15 -> row r ; lanes 16-31 -> row r+8.
    float sc3[8], sc2[8];
#pragma unroll
    for (int r = 0; r < 8; ++r) {
        float es = __shfl(entsq, r + 8 * hi, 32);
        sc3[r] = expf(-0.5f * (m3[r] + es));
        sc2[r] = expf(-0.5f * (m2[r] + es));
    }

    long long nbe = nb_entities[b];
    if (lo == 0) {                             // one writer per 16-lane half
        float* out_sp = out;
        float* out_po = out + (size_t)B * N;
#pragma unroll
        for (int r = 0; r < 8; ++r) {
            int row = rowBase + r + 8 * hi;
            bool valid = (long long)row < nbe;
            out_sp[(size_t)b * N + row] = valid ? sc3[r] : 0.f;
            out_po[(size_t)b * N + row] = valid ? sc2[r] : 0.f;
        }
    }
}

// ---------------------------------------------------------------------------
extern "C" void kernel_launch(void* const* d_in, const int* in_sizes, int n_in,
                              void* d_out, int out_size, void* d_ws, size_t ws_size,
                              hipStream_t stream)
{
    const float* rel  = (const float*)d_in[0];
    const float* arg1 = (const float*)d_in[1];
    const float* arg2 = (const float*)d_in[2];
    const float* f1   = (const float*)d_in[3];
    const float* f2   = (const float*)d_in[4];
    const float* f3   = (const float*)d_in[5];
    const float* ent  = (const float*)d_in[6];
    const long long* nb_facts    = (const long long*)d_in[7];
    const long long* nb_entities = (const long long*)d_in[8];
    float* out = (float*)d_out;

    // Derive shapes from input sizes (reference: B=8, N=4096, F=2048, E=128)
    int B = in_sizes[7];
    int E = in_sizes[0] / B;
    int F = in_sizes[3] / (B * E);
    int N = in_sizes[6] / (B * E);

    float* cfix3 = (float*)d_ws;                 // [B,F]
    float* cfix2 = cfix3 + (size_t)B * F;        // [B,F]

    int prepass_blocks = (B * F + 255) / 256;
    cfix_kernel<<<prepass_blocks, 256, 0, stream>>>(
        rel, arg1, arg2, f1, f2, f3, nb_facts, cfix3, cfix2, B, F, E);

    int main_blocks = B * (N / 128);
    score_kernel<<<main_blocks, 256, 0, stream>>>(
        ent, f2, f3, cfix3, cfix2, nb_facts, nb_entities, out, B, N, F, E);
}